// Net_85186381349135
// MI455X (gfx1250) — compile-verified
//
#include <hip/hip_runtime.h>
#include <hip/hip_bf16.h>

typedef __attribute__((ext_vector_type(2))) float v2f;
typedef __attribute__((ext_vector_type(8))) float v8f;

// ---------------------------------------------------------------------------
// 0) init: deg = 1.0 (self-loop), pool = 0, cnt = 0
// ---------------------------------------------------------------------------
__global__ void gcn_init_kernel(float* __restrict__ deg,
                                float* __restrict__ pool,
                                float* __restrict__ cnt,
                                int N, int G) {
    int t = blockIdx.x * blockDim.x + threadIdx.x;
    if (t < N) deg[t] = 1.0f;          // self-loop contributes 1 to every degree
    if (t < G * 16) pool[t] = 0.0f;
    if (t < G) cnt[t] = 0.0f;
}

// ---------------------------------------------------------------------------
// 1) degree: deg[dst] += 1 per edge
// ---------------------------------------------------------------------------
__global__ void gcn_deg_kernel(const long long* __restrict__ ei,
                               float* __restrict__ deg, int E) {
    int e = blockIdx.x * blockDim.x + threadIdx.x;
    if (e < E) {
        int d = (int)ei[(long long)E + e];   // edge_index[1][e]
        atomicAdd(&deg[d], 1.0f);
    }
}

// ---------------------------------------------------------------------------
// 2) dinv = rsqrt(deg)   (deg >= 1 always due to self-loop)
// ---------------------------------------------------------------------------
__global__ void gcn_dinv_kernel(const float* __restrict__ deg,
                                float* __restrict__ dinv, int N) {
    int n = blockIdx.x * blockDim.x + threadIdx.x;
    if (n < N) dinv[n] = rsqrtf(deg[n]);
}

// ---------------------------------------------------------------------------
// 3) WMMA GEMM: hs = (x @ W1) * dinv[row]; acc seeded with hs (self-loop term)
//    One wave per 16-row tile. K=128 -> 32 x V_WMMA_F32_16X16X4_F32.
//    A 16x4 f32 layout: lanes 0-15: {K=k,k+1}; lanes 16-31: {K=k+2,k+3}.
//    B 4x16 f32 layout mirrored. C/D: VGPR r = row (r | r+8), col = lane&15.
// ---------------------------------------------------------------------------
__global__ void gcn_gemm_wmma_kernel(const float* __restrict__ x,
                                     const float* __restrict__ W1,
                                     const float* __restrict__ dinv,
                                     float* __restrict__ hs,
                                     float* __restrict__ acc,
                                     int nTiles) {
    int wavesPerBlock = blockDim.x >> 5;
    int tile = blockIdx.x * wavesPerBlock + ((int)threadIdx.x >> 5);
    if (tile >= nTiles) return;                 // wave-uniform: EXEC stays all-1s
    int lane = (int)threadIdx.x & 31;
    int m    = lane & 15;                       // A-row / B-col
    int koff = (lane >> 4) << 1;                // 0 or 2
    const float* xrow = x + ((long long)(tile * 16 + m)) * 128;

    v8f c = {};
#pragma unroll
    for (int k = 0; k < 128; k += 4) {
        v2f a, b;
        a.x = xrow[k + koff];
        a.y = xrow[k + koff + 1];
        b.x = W1[(k + koff) * 16 + m];
        b.y = W1[(k + koff + 1) * 16 + m];
        c = __builtin_amdgcn_wmma_f32_16x16x4_f32(
                /*neg_a=*/false, a, /*neg_b=*/false, b,
                /*c_mod=*/(short)0, c, /*reuse_a=*/false, /*reuse_b=*/false);
    }

    // Epilogue: scale row r by dinv[row], write hs and seed acc (self-loop)
    int rbase = tile * 16 + ((lane >> 4) << 3);   // +0 (lanes 0-15) or +8
#pragma unroll
    for (int r = 0; r < 8; ++r) {
        int row = rbase + r;
        float v = c[r] * dinv[row];
        hs[(long long)row * 16 + m]  = v;
        acc[(long long)row * 16 + m] = v;
    }
}

// ---------------------------------------------------------------------------
// 4) edge scatter: acc[dst][j] += hs[src][j], 16 lanes per edge
// ---------------------------------------------------------------------------
__global__ void gcn_scatter_kernel(const long long* __restrict__ ei,
                                   const float* __restrict__ hs,
                                   float* __restrict__ acc, int E) {
    int t = blockIdx.x * blockDim.x + threadIdx.x;
    int e = t >> 4;
    if (e < E) {
        int j = t & 15;
        int s = (int)ei[e];                     // edge_index[0][e]
        int d = (int)ei[(long long)E + e];      // edge_index[1][e]
        atomicAdd(&acc[(long long)d * 16 + j], hs[(long long)s * 16 + j]);
    }
}

// ---------------------------------------------------------------------------
// 5) node finalize + mean-pool accumulate:
//    v = relu(acc*dinv + b1); pool[batch[n]] += v; cnt[batch[n]] += 1
// ---------------------------------------------------------------------------
__global__ void gcn_pool_kernel(const float* __restrict__ acc,
                                const float* __restrict__ dinv,
                                const float* __restrict__ b1,
                                const long long* __restrict__ batch,
                                float* __restrict__ pool,
                                float* __restrict__ cnt, int N) {
    int t = blockIdx.x * blockDim.x + threadIdx.x;
    int n = t >> 4;
    if (n < N) {
        int j = t & 15;
        float v = fmaxf(acc[(long long)n * 16 + j] * dinv[n] + b1[j], 0.0f);
        int g = (int)batch[n];
        atomicAdd(&pool[g * 16 + j], v);
        if (j == 0) atomicAdd(&cnt[g], 1.0f);
    }
}

// ---------------------------------------------------------------------------
// 6) final WMMA GEMM: out = (pool/cnt) @ W2 + b2   [G x 10], N padded to 16
// ---------------------------------------------------------------------------
__global__ void gcn_final_wmma_kernel(const float* __restrict__ pool,
                                      const float* __restrict__ cnt,
                                      const float* __restrict__ W2,
                                      const float* __restrict__ b2,
                                      float* __restrict__ out, int nTiles) {
    int wavesPerBlock = blockDim.x >> 5;
    int tile = blockIdx.x * wavesPerBlock + ((int)threadIdx.x >> 5);
    if (tile >= nTiles) return;                 // wave-uniform
    int lane = (int)threadIdx.x & 31;
    int m    = lane & 15;
    int koff = (lane >> 4) << 1;
    int grow = tile * 16 + m;
    float inv = 1.0f / fmaxf(cnt[grow], 1.0f);

    v8f c = {};
#pragma unroll
    for (int k = 0; k < 16; k += 4) {
        int k0 = k + koff;
        v2f a, b;
        a.x = pool[grow * 16 + k0] * inv;
        a.y = pool[grow * 16 + k0 + 1] * inv;
        b.x = (m < 10) ? W2[k0 * 10 + m] : 0.0f;        // zero-pad cols 10..15
        b.y = (m < 10) ? W2[(k0 + 1) * 10 + m] : 0.0f;
        c = __builtin_amdgcn_wmma_f32_16x16x4_f32(
                false, a, false, b, (short)0, c, false, false);
    }

    if (m < 10) {
        float bb = b2[m];
        int rbase = tile * 16 + ((lane >> 4) << 3);
#pragma unroll
        for (int r = 0; r < 8; ++r)
            out[(rbase + r) * 10 + m] = c[r] + bb;
    }
}

// ---------------------------------------------------------------------------
// host launcher
// ---------------------------------------------------------------------------
extern "C" void kernel_launch(void* const* d_in, const int* in_sizes, int n_in,
                              void* d_out, int out_size, void* d_ws, size_t ws_size,
                              hipStream_t stream) {
    const float*     x     = (const float*)d_in[0];       // [N,128]
    const long long* ei    = (const long long*)d_in[1];   // [2,E] int64
    const long long* batch = (const long long*)d_in[2];   // [N] int64
    const float*     W1    = (const float*)d_in[3];       // [128,16]
    const float*     b1    = (const float*)d_in[4];       // [16]
    const float*     W2    = (const float*)d_in[5];       // [16,10]
    const float*     b2    = (const float*)d_in[6];       // [10]
    float*           out   = (float*)d_out;               // [G,10]

    const int N = in_sizes[0] / 128;
    const int E = in_sizes[1] / 2;
    const int G = out_size / 10;

    // workspace layout (floats): deg[N] | dinv[N] | hs[N*16] | acc[N*16] | pool[G*16] | cnt[G]
    float* ws   = (float*)d_ws;
    float* deg  = ws;
    float* dinv = deg  + N;
    float* hs   = dinv + N;
    float* acc  = hs   + (size_t)N * 16;
    float* pool = acc  + (size_t)N * 16;
    float* cnt  = pool + (size_t)G * 16;

    const int B = 256;
    int initM = N > G * 16 ? N : G * 16;

    gcn_init_kernel<<<(initM + B - 1) / B, B, 0, stream>>>(deg, pool, cnt, N, G);
    gcn_deg_kernel<<<(E + B - 1) / B, B, 0, stream>>>(ei, deg, E);
    gcn_dinv_kernel<<<(N + B - 1) / B, B, 0, stream>>>(deg, dinv, N);

    int nTiles = (N + 15) / 16;                  // 6250 (N is a multiple of 16)
    int wavesPerBlock = 4;                        // 128 threads = 4 wave32
    int gemmBlocks = (nTiles + wavesPerBlock - 1) / wavesPerBlock;
    gcn_gemm_wmma_kernel<<<gemmBlocks, wavesPerBlock * 32, 0, stream>>>(
        x, W1, dinv, hs, acc, nTiles);

    long long scatterThreads = (long long)E * 16;
    gcn_scatter_kernel<<<(int)((scatterThreads + B - 1) / B), B, 0, stream>>>(
        ei, hs, acc, E);

    long long poolThreads = (long long)N * 16;
    gcn_pool_kernel<<<(int)((poolThreads + B - 1) / B), B, 0, stream>>>(
        acc, dinv, b1, batch, pool, cnt, N);

    int fTiles = (G + 15) / 16;                  // 64
    int fBlocks = (fTiles + wavesPerBlock - 1) / wavesPerBlock;
    gcn_final_wmma_kernel<<<fBlocks, wavesPerBlock * 32, 0, stream>>>(
        pool, cnt, W2, b2, out, fTiles);
}